// RANet_45964740001820
// MI455X (gfx1250) — compile-verified
//
#include <hip/hip_runtime.h>
#include <hip/hip_bf16.h>

typedef __attribute__((ext_vector_type(2))) float v2f;
typedef __attribute__((ext_vector_type(8))) float v8f;

#define PROBS_PER_BLOCK 16
#define H_STRIDE 516   // 512 + 4 pad -> rows land on distinct banks for A-frag reads

__global__ __launch_bounds__(128) void ranet_fused_kernel(
    const float* __restrict__ gxy,   // (P, 32, 7)
    const float* __restrict__ w1,    // (32, 3)
    const float* __restrict__ b1,    // (32)
    const float* __restrict__ g1,    // (32)
    const float* __restrict__ be1,   // (32)
    const float* __restrict__ w2,    // (64, 512)  [o][c*16 + h*4 + w]
    const float* __restrict__ b2,    // (64)
    const float* __restrict__ g2,    // (64)
    const float* __restrict__ be2,   // (64)
    float* __restrict__ out,         // (P, 64)
    int P)
{
    __shared__ float sBins[PROBS_PER_BLOCK * 48];       // [p][{cnt,rcs,vr}][16]
    __shared__ float sH[PROBS_PER_BLOCK * H_STRIDE];    // [p][k=c*16+px], padded

    const int lane  = threadIdx.x & 31;
    const int wave  = threadIdx.x >> 5;     // 0..3
    const int pbase = blockIdx.x * PROBS_PER_BLOCK;

    // ---------------- Phase 1: ra-map (lane = point) ----------------
    for (int it = 0; it < 4; ++it) {
        const int pl = wave * 4 + it;
        const int p  = pbase + pl;
        float* bins = &sBins[pl * 48];
        if (lane < 16) {            // count / rcs_max / vr_max all clamp at 0
            bins[lane] = 0.0f; bins[16 + lane] = 0.0f; bins[32 + lane] = 0.0f;
        }
        if (p < P) {
            const float* pt = gxy + ((size_t)p * 32 + lane) * 7;
            const float x = pt[0], y = pt[1], rcs = pt[3], vr = pt[5];
            const float r = sqrtf(x * x + y * y);
            const float a = atan2f(y, x);
            float rlo = r, rhi = r, alo = a, ahi = a;
            #pragma unroll
            for (int m = 16; m >= 1; m >>= 1) {
                rlo = fminf(rlo, __shfl_xor(rlo, m, 32));
                rhi = fmaxf(rhi, __shfl_xor(rhi, m, 32));
                alo = fminf(alo, __shfl_xor(alo, m, 32));
                ahi = fmaxf(ahi, __shfl_xor(ahi, m, 32));
            }
            float ur = (rhi - rlo) * 0.25f; if (ur == 0.0f) ur = 1.0f;
            float ua = (ahi - alo) * 0.25f; if (ua == 0.0f) ua = 1.0f;
            int ri = (int)floorf((r - rlo) / ur); ri = ri > 3 ? 3 : (ri < 0 ? 0 : ri);
            int ai = (int)floorf((a - alo) / ua); ai = ai > 3 ? 3 : (ai < 0 ? 0 : ai);
            const int bin = ri * 4 + ai;
            atomicAdd(&bins[bin], 1.0f);                               // ds_add_f32
            // signed-int max == float max here: negatives lose to the 0.0 floor
            atomicMax((int*)&bins[16 + bin], __float_as_int(rcs));     // ds_max_i32
            atomicMax((int*)&bins[32 + bin], __float_as_int(vr));
        }
    }

    // ---------------- Phase 2: conv1 + GN1 + ReLU (lane = channel) ----------------
    {
        const int c = lane;
        const float w10 = w1[c * 3 + 0], w11 = w1[c * 3 + 1], w12 = w1[c * 3 + 2];
        const float bb1 = b1[c], gg1 = g1[c], bbe1 = be1[c];
        for (int it = 0; it < 4; ++it) {
            const int pl = wave * 4 + it;
            const float* bins = &sBins[pl * 48];
            float t[16];
            float s = 0.0f, sq = 0.0f;
            #pragma unroll
            for (int px = 0; px < 16; ++px) {
                const float v = w10 * bins[px] + w11 * bins[16 + px] + w12 * bins[32 + px] + bb1;
                t[px] = v; s += v; sq += v * v;
            }
            // GN1: group = 4 adjacent channels (lanes) x 16 px = 64 elems
            s  += __shfl_xor(s, 1, 32);  s  += __shfl_xor(s, 2, 32);
            sq += __shfl_xor(sq, 1, 32); sq += __shfl_xor(sq, 2, 32);
            const float mean = s * (1.0f / 64.0f);
            const float var  = sq * (1.0f / 64.0f) - mean * mean;
            const float rs   = rsqrtf(var + 1e-5f);
            float* hrow = &sH[pl * H_STRIDE + c * 16];
            #pragma unroll
            for (int px = 0; px < 16; ++px)
                hrow[px] = fmaxf(0.0f, (t[px] - mean) * rs * gg1 + bbe1);
        }
    }

    __syncthreads();

    // ---------------- Phase 3: conv2 as WMMA f32 16x16x4 GEMM ----------------
    // D(16 probs x 16 chans per wave) = H(16x512) * W2^T(512x16)
    // A frag: lane half h = lane>>4; VGPR j holds K = k + 2h + j, M = lane&15
    // B frag: same K indexing, N = lane&15
    const int nIdx    = lane & 15;
    const int kk      = (lane >> 4) * 2;
    const int colBase = wave * 16;
    const float* w2c  = w2 + (size_t)(colBase + nIdx) * 512;
    const float* hA   = &sH[nIdx * H_STRIDE];   // A row M = lane&15

    v8f acc = (v8f){0.f, 0.f, 0.f, 0.f, 0.f, 0.f, 0.f, 0.f};
    #pragma unroll 8
    for (int k = 0; k < 512; k += 4) {
        v2f Af = { hA[k + kk],  hA[k + kk + 1]  };
        v2f Bf = { w2c[k + kk], w2c[k + kk + 1] };
        acc = __builtin_amdgcn_wmma_f32_16x16x4_f32(
            /*neg_a=*/false, Af, /*neg_b=*/false, Bf,
            /*c_mod=*/(short)0, acc, /*reuse_a=*/false, /*reuse_b=*/false);
    }

    // ---------------- Epilogue: bias + GN2 (8-ch groups) + ReLU + store ----------------
    {
        const int ch = colBase + nIdx;            // 0..63
        const float bb2 = b2[ch], gg2 = g2[ch], bbe2 = be2[ch];
        const int half = lane >> 4;
        #pragma unroll
        for (int r = 0; r < 8; ++r) {
            const float v = acc[r] + bb2;
            float s = v, sq = v * v;
            // group = 8 adjacent channels = 8 adjacent lanes (xor 1,2,4 stays in-group)
            s  += __shfl_xor(s, 1, 32);  s  += __shfl_xor(s, 2, 32);  s  += __shfl_xor(s, 4, 32);
            sq += __shfl_xor(sq, 1, 32); sq += __shfl_xor(sq, 2, 32); sq += __shfl_xor(sq, 4, 32);
            const float mean = s * 0.125f;
            const float var  = sq * 0.125f - mean * mean;
            const float o = fmaxf(0.0f, (v - mean) * rsqrtf(var + 1e-5f) * gg2 + bbe2);
            const int p = pbase + r + 8 * half;   // D-frag row: VGPR r -> M = r + 8*(lane>=16)
            if (p < P) out[(size_t)p * 64 + ch] = o;
        }
    }
}

extern "C" void kernel_launch(void* const* d_in, const int* in_sizes, int n_in,
                              void* d_out, int out_size, void* d_ws, size_t ws_size,
                              hipStream_t stream) {
    const float* gxy = (const float*)d_in[0];
    const float* w1  = (const float*)d_in[1];
    const float* b1  = (const float*)d_in[2];
    const float* g1  = (const float*)d_in[3];
    const float* be1 = (const float*)d_in[4];
    const float* w2  = (const float*)d_in[5];
    const float* b2  = (const float*)d_in[6];
    const float* g2  = (const float*)d_in[7];
    const float* be2 = (const float*)d_in[8];
    float* out = (float*)d_out;

    const int P = in_sizes[0] / (32 * 7);            // B*M problems
    const int blocks = (P + PROBS_PER_BLOCK - 1) / PROBS_PER_BLOCK;

    ranet_fused_kernel<<<blocks, 128, 0, stream>>>(
        gxy, w1, b1, g1, be1, w2, b2, g2, be2, out, P);
}